// PillarEncoder_84310208020617
// MI455X (gfx1250) — compile-verified
//
#include <hip/hip_runtime.h>
#include <hip/hip_bf16.h>

typedef __attribute__((ext_vector_type(2))) float v2f;
typedef __attribute__((ext_vector_type(8))) float v8f;

#define X_L 432
#define Y_L 496
#define IMG_H 96
#define IMG_W 320
#define C_OUT 64

// ---------------------------------------------------------------------------
// Zero-fill the two output canvases (220 MB -> pure HBM store bandwidth).
// ---------------------------------------------------------------------------
__global__ void zero_kernel(float* __restrict__ out, long long n) {
    long long i = ((long long)blockIdx.x * blockDim.x + threadIdx.x) * 4;
    const long long stride = (long long)gridDim.x * blockDim.x * 4;
    const float4 z = make_float4(0.f, 0.f, 0.f, 0.f);
    for (; i + 3 < n; i += stride) *(float4*)(out + i) = z;
    const long long tid = (long long)blockIdx.x * blockDim.x + threadIdx.x;
    const long long tail = n & 3LL;
    if (tid < tail) out[n - 1 - tid] = 0.f;
}

// ---------------------------------------------------------------------------
// One wave32 per pillar. lane == point index (N == 32).
//  - feats(32x9) @ conv_w^T(9x64) via V_WMMA_F32_16X16X4_F32
//    (2 M-tiles x 4 N-tiles x 3 K-steps, K padded 9->12)
//  - BN/ReLU after the row-max (BN is strictly increasing: gamma > 0)
//  - top-5 nearest points -> project -> bilinear sample -> average
// ---------------------------------------------------------------------------
__global__ __launch_bounds__(256) void pillar_kernel(
    const float4* __restrict__ pillars,   // (P,32,4)
    const int*    __restrict__ coors,     // (P,4)
    const int*    __restrict__ npoints,   // (P,)
    const float*  __restrict__ img,       // (B,64,96,320)
    const int*    __restrict__ ishapes,   // (B,2)
    const float*  __restrict__ projm,     // (B,3,4)
    const float*  __restrict__ convw,     // (64,9)
    const float*  __restrict__ bgam,
    const float*  __restrict__ bbeta,
    const float*  __restrict__ bmean,
    const float*  __restrict__ bvar,
    float* __restrict__ out_l,            // (B,64,496,432)
    float* __restrict__ out_i,            // (B,64,496,432)
    int P)
{
    const int lane = threadIdx.x & 31;
    const int wid  = (int)(((long long)blockIdx.x * blockDim.x + threadIdx.x) >> 5);
    if (wid >= P) return;                 // wave-uniform: EXEC stays all-1s
    const int p = wid;

    // ---- load this lane's point -------------------------------------------
    const float4 pt = pillars[p * 32 + lane];
    const int  np_  = npoints[p];
    const bool valid = lane < np_;

    // centroid: reference sums ALL 32 points, divides by npoints
    float sx = pt.x, sy = pt.y, sz = pt.z;
    #pragma unroll
    for (int m = 16; m >= 1; m >>= 1) {
        sx += __shfl_xor(sx, m, 32);
        sy += __shfl_xor(sy, m, 32);
        sz += __shfl_xor(sz, m, 32);
    }
    const float inv_np = 1.0f / (float)np_;
    const float mx = sx * inv_np, my = sy * inv_np, mz = sz * inv_np;

    const int b  = coors[p * 4 + 0];
    const int xc = coors[p * 4 + 1];
    const int yc = coors[p * 4 + 2];
    const float cx = (float)xc * 0.16f + 0.08f;
    const float cy = (float)yc * 0.16f + (0.08f - 39.68f);

    const float xo = pt.x - cx, yo = pt.y - cy;
    const float ox = pt.x - mx, oy = pt.y - my, oz = pt.z - mz;

    float f[12];
    f[0] = xo; f[1] = yo; f[2] = pt.z; f[3] = pt.w;
    f[4] = ox; f[5] = oy; f[6] = oz;   f[7] = xo; f[8] = yo;
    f[9] = 0.f; f[10] = 0.f; f[11] = 0.f;
    const float msk = valid ? 1.0f : 0.0f;
    #pragma unroll
    for (int i = 0; i < 9; ++i) f[i] *= msk;

    const float dist = valid ? (ox * ox + oy * oy + oz * oz) : 1.0e6f;

    // ---- B tiles: B[k][n] = conv_w[n][k]; 4x16 layout: VGPR r -> K=r,K=r+2 -
    const int lh = lane & 15;
    const int hi = lane >> 4;             // 0: lanes 0-15, 1: lanes 16-31
    v2f Bt[3][4];
    #pragma unroll
    for (int nt = 0; nt < 4; ++nt) {
        const int n = nt * 16 + lh;
        #pragma unroll
        for (int kci = 0; kci < 3; ++kci) {
            const int k0 = kci * 4 + hi * 2;
            const int k1 = k0 + 1;
            Bt[kci][nt].x = (k0 < 9) ? convw[n * 9 + k0] : 0.0f;
            Bt[kci][nt].y = (k1 < 9) ? convw[n * 9 + k1] : 0.0f;
        }
    }

    // ---- A tiles: 16x4 layout: lane<16 K={r}, lane>=16 K={r+2} -------------
    float ga[12], gb[12];
    #pragma unroll
    for (int i = 0; i < 9; ++i) {
        ga[i] = __shfl(f[i], lh, 32);        // M-tile 0: points 0-15
        gb[i] = __shfl(f[i], lh + 16, 32);   // M-tile 1: points 16-31
    }
    ga[9] = ga[10] = ga[11] = 0.f;
    gb[9] = gb[10] = gb[11] = 0.f;

    v2f A[2][3];
    #pragma unroll
    for (int kci = 0; kci < 3; ++kci) {
        const int kb = kci * 4;
        A[0][kci].x = hi ? ga[kb + 2] : ga[kb + 0];
        A[0][kci].y = hi ? ga[kb + 3] : ga[kb + 1];
        A[1][kci].x = hi ? gb[kb + 2] : gb[kb + 0];
        A[1][kci].y = hi ? gb[kb + 3] : gb[kb + 1];
    }

    // ---- 24x V_WMMA_F32_16X16X4_F32 ---------------------------------------
    v8f acc[2][4];
    #pragma unroll
    for (int mt = 0; mt < 2; ++mt) {
        #pragma unroll
        for (int nt = 0; nt < 4; ++nt) {
            v8f c = {};
            #pragma unroll
            for (int kci = 0; kci < 3; ++kci) {
                c = __builtin_amdgcn_wmma_f32_16x16x4_f32(
                        false, A[mt][kci], false, Bt[kci][nt],
                        (short)0, c, false, false);
            }
            acc[mt][nt] = c;
        }
    }

    // ---- max over 32 points, then BN + ReLU (monotone), scatter -----------
    #pragma unroll
    for (int nt = 0; nt < 4; ++nt) {
        float m0 = acc[0][nt][0];
        float m1 = acc[1][nt][0];
        #pragma unroll
        for (int i = 1; i < 8; ++i) {
            m0 = fmaxf(m0, acc[0][nt][i]);
            m1 = fmaxf(m1, acc[1][nt][i]);
        }
        float mm = fmaxf(m0, m1);
        mm = fmaxf(mm, __shfl_xor(mm, 16, 32));
        if (lane < 16) {
            const int n = nt * 16 + lane;
            const float sc  = bgam[n] * rsqrtf(bvar[n] + 1.0e-3f);
            const float val = (mm - bmean[n]) * sc + bbeta[n];
            out_l[(((long long)b * C_OUT + n) * Y_L + yc) * X_L + xc] =
                fmaxf(val, 0.0f);
        }
    }

    // ---- top-5 nearest points + bilinear image sampling -------------------
    const float* M = projm + b * 12;
    const float M0 = M[0], M1 = M[1], M2 = M[2],  M3  = M[3];
    const float M4 = M[4], M5 = M[5], M6 = M[6],  M7  = M[7];
    const float M8 = M[8], M9 = M[9], Ma = M[10], Mb_ = M[11];
    const float su = (float)IMG_W / (float)ishapes[b * 2 + 1];
    const float sv = (float)IMG_H / (float)ishapes[b * 2 + 0];

    const int c0 = lane, c1 = lane + 32;   // 2 channels per lane
    const float* bp0 = img + ((long long)b * C_OUT + c0) * (IMG_H * IMG_W);
    const float* bp1 = img + ((long long)b * C_OUT + c1) * (IMG_H * IMG_W);

    float d = dist;
    float sum0 = 0.f, sum1 = 0.f, cnt = 0.f;
    #pragma unroll
    for (int it = 0; it < 5; ++it) {
        // wave arg-min with index tie-break (matches stable top_k)
        float bd = d; int bi = lane;
        #pragma unroll
        for (int m = 16; m >= 1; m >>= 1) {
            const float od = __shfl_xor(bd, m, 32);
            const int   oi = __shfl_xor(bi, m, 32);
            if (od < bd || (od == bd && oi < bi)) { bd = od; bi = oi; }
        }
        if (lane == bi) d = 3.0e38f;       // remove winner for next pass

        const float tx = __shfl(pt.x, bi, 32);
        const float ty = __shfl(pt.y, bi, 32);
        const float tz = __shfl(pt.z, bi, 32);
        const bool tvalid = bi < np_;

        const float pw = M8 * tx + M9 * ty + Ma * tz + Mb_;
        const float iw = 1.0f / pw;
        const float us = (M0 * tx + M1 * ty + M2 * tz + M3) * iw * su;
        const float vs = (M4 * tx + M5 * ty + M6 * tz + M7) * iw * sv;

        // wave-uniform condition (bi, us, vs uniform across lanes)
        if (tvalid && us >= 0.f && us <= (float)(IMG_W - 1) &&
                      vs >= 0.f && vs <= (float)(IMG_H - 1)) {
            const float uc_ = fminf(fmaxf(us, 0.f), (float)(IMG_W - 1));
            const float vc_ = fminf(fmaxf(vs, 0.f), (float)(IMG_H - 1));
            const int x0 = (int)floorf(uc_), y0 = (int)floorf(vc_);
            const int x1 = min(x0 + 1, IMG_W - 1);
            const int y1 = min(y0 + 1, IMG_H - 1);
            const float wx = uc_ - (float)x0, wy = vc_ - (float)y0;
            const float w00 = (1.f - wx) * (1.f - wy);
            const float w01 = wx * (1.f - wy);
            const float w10 = (1.f - wx) * wy;
            const float w11 = wx * wy;
            const int i00 = y0 * IMG_W + x0, i01 = y0 * IMG_W + x1;
            const int i10 = y1 * IMG_W + x0, i11 = y1 * IMG_W + x1;
            sum0 += w00 * bp0[i00] + w01 * bp0[i01] + w10 * bp0[i10] + w11 * bp0[i11];
            sum1 += w00 * bp1[i00] + w01 * bp1[i01] + w10 * bp1[i10] + w11 * bp1[i11];
            cnt  += 1.0f;
        }
    }
    const float inv = 1.0f / fmaxf(cnt, 1.0f);
    out_i[(((long long)b * C_OUT + c0) * Y_L + yc) * X_L + xc] = sum0 * inv;
    out_i[(((long long)b * C_OUT + c1) * Y_L + yc) * X_L + xc] = sum1 * inv;
}

extern "C" void kernel_launch(void* const* d_in, const int* in_sizes, int n_in,
                              void* d_out, int out_size, void* d_ws, size_t ws_size,
                              hipStream_t stream) {
    const float4* pillars = (const float4*)d_in[0];
    const int*    coors   = (const int*)d_in[1];
    const int*    npts    = (const int*)d_in[2];
    const float*  img     = (const float*)d_in[3];
    const int*    ishapes = (const int*)d_in[4];
    const float*  projm   = (const float*)d_in[5];
    const float*  convw   = (const float*)d_in[6];
    const float*  bgam    = (const float*)d_in[7];
    const float*  bbeta   = (const float*)d_in[8];
    const float*  bmean   = (const float*)d_in[9];
    const float*  bvar    = (const float*)d_in[10];

    const int P = in_sizes[2];                       // 20000
    float* out = (float*)d_out;
    const long long half = (long long)2 * C_OUT * Y_L * X_L;  // 27,426,816
    float* out_l = out;
    float* out_i = out + half;

    zero_kernel<<<4096, 256, 0, stream>>>(out, (long long)out_size);

    const int blocks = (P + 7) / 8;                  // 8 waves (pillars) per block
    pillar_kernel<<<blocks, 256, 0, stream>>>(
        pillars, coors, npts, img, ishapes, projm, convw,
        bgam, bbeta, bmean, bvar, out_l, out_i, P);
}